// StatisticalFeatureLoss_14783277433470
// MI455X (gfx1250) — compile-verified
//
#include <hip/hip_runtime.h>
#include <hip/hip_bf16.h>

typedef __attribute__((ext_vector_type(2))) float v2f;
typedef __attribute__((ext_vector_type(8))) float v8f;

#define EPSF 1e-8f
#define IMG 512
#define NPLANES 48          // 16 batch * 3 channels
#define WAVES_PER_BLOCK 4
#define TILES_PER_PLANE 1024    // (512/16)^2
#define TOTAL_TILES (NPLANES * TILES_PER_PLANE)       // 49152
#define BLOCKS_PER_K (TOTAL_TILES / WAVES_PER_BLOCK)  // 12288

// Fully fused statistical-feature kernel for one Gaussian window size K.
// One wave32 owns one 16x16 output tile; the vertical conv passes for
// E[x^2], m3, m4 run on the matrix pipe as banded 16x16xKD matmuls chunked
// into V_WMMA_F32_16X16X4_F32 ops.
template <int K>
__global__ __launch_bounds__(128) void sfl_kernel(
    const float* __restrict__ pred, const float* __restrict__ tgt,
    float* __restrict__ partial) {
  constexpr int P   = K / 2;        // conv pad
  constexpr int H   = 2 * P;        // two-level halo
  constexpr int XS  = 16 + 2 * H;   // raw input region side
  constexpr int MS  = 16 + 2 * P;   // mean region side == WMMA K-dim (KD)
  constexpr int XSP = XS + 1;       // padded LDS strides (bank-friendly)
  constexpr int MSP = MS + 1;
  constexpr int NCH = (MS + 3) / 4; // number of 16x16x4 WMMA chunks
  constexpr int NK  = NCH * 4;      // padded K-dim (multiple of 4)
  // Tb must hold rows up to P + NK - 1 (pad rows are zeroed once, so the
  // WMMA B gathers can be unconditional ds_loads).
  constexpr int TROWS = (P + NK > XS) ? (P + NK) : XS;

  __shared__ float Xb[WAVES_PER_BLOCK][XS * XSP];     // raw tile + halo
  __shared__ float Tb[WAVES_PER_BLOCK][TROWS * MSP];  // horizontal-pass temp
  __shared__ float Mb[WAVES_PER_BLOCK][MS * MSP];     // mean over halo region
  __shared__ float Pb[WAVES_PER_BLOCK][MS * MSP];     // (x-mean)^3 / ^4
  __shared__ float Ab[WAVES_PER_BLOCK][32 * NCH * 2]; // per-lane A fragments
  __shared__ float wsum[WAVES_PER_BLOCK];

  const int tid  = threadIdx.x;
  const int w    = tid >> 5;   // wave id within block
  const int lane = tid & 31;
  const int hi   = lane >> 4;  // WMMA half-wave selector
  const int lo   = lane & 15;  // WMMA row/col within half

  // Normalized 1D Gaussian, k/6 sigma (matches reference). Register-resident
  // (compile-time indices only).
  float greg[K];
  {
    const float sigma = (float)K / 6.0f;
    float s = 0.0f;
#pragma unroll
    for (int j = 0; j < K; ++j) {
      float c = (float)j - (float)(K / 2);
      greg[j] = __expf(-c * c / (2.0f * sigma * sigma));
      s += greg[j];
    }
    const float inv = 1.0f / s;
#pragma unroll
    for (int j = 0; j < K; ++j) greg[j] *= inv;
  }

  // Precompute all WMMA A fragments (banded Gaussian, 16xNK) ONCE and park
  // them in LDS so the compiler cannot rematerialize the cndmask chain at
  // every conv pass. A layout per ISA: lane L holds A[L&15][ch*4+2*(L>>4)+j]
  // in VGPR j. Band value Gband[m][t] = g[t-m] for 0<=t-m<K, else 0
  // (analytically zero for all t >= MS).
  {
    float* ab = &Ab[w][lane * (NCH * 2)];
#pragma unroll
    for (int ch = 0; ch < NCH; ++ch) {
#pragma unroll
      for (int j = 0; j < 2; ++j) {
        const int d = ch * 4 + 2 * hi + j - lo;
        float av = 0.0f;
#pragma unroll
        for (int q = 0; q < K; ++q) av = (d == q) ? greg[q] : av;
        ab[ch * 2 + j] = av;
      }
    }
  }

  // Tile coordinates for this wave.
  const int tile  = blockIdx.x * WAVES_PER_BLOCK + w;
  const int plane = tile >> 10;
  const int tpos  = tile & 1023;
  const int gr0   = (tpos >> 5) << 4;
  const int gc0   = (tpos & 31) << 4;
  const size_t pbase = (size_t)plane * IMG * IMG;

  // Zero the K-dim pad rows of Tb once (K=3/K=7 have NK > MS).
  if constexpr (NK > MS) {
    for (int idx = lane; idx < (NK - MS) * 16; idx += 32) {
      const int t = MS + (idx >> 4), n = idx & 15;
      Tb[w][(P + t) * MSP + n] = 0.0f;
    }
  }

  // Banded vertical conv on the matrix pipe:
  //   D(16x16) = Gband(16xNK) x Tmp(NKx16), chunked by 4 along K.
  // Gather A (ds_load_b64 per chunk) and all B fragments first, then chain
  // the WMMAs back-to-back.
  auto conv_v = [&](const float* tb) -> v8f {
    const float* ab = &Ab[w][lane * (NCH * 2)];
    v2f aF[NCH], bF[NCH];
#pragma unroll
    for (int ch = 0; ch < NCH; ++ch) {
      aF[ch] = *(const v2f*)(ab + ch * 2);
#pragma unroll
      for (int j = 0; j < 2; ++j) {
        const int t = ch * 4 + 2 * hi + j;
        bF[ch][j] = tb[(P + t) * MSP + lo];
      }
    }
    v8f c = {0.f, 0.f, 0.f, 0.f, 0.f, 0.f, 0.f, 0.f};
#pragma unroll
    for (int ch = 0; ch < NCH; ++ch) {
      // 8 args: (neg_a, A, neg_b, B, c_mod, C, reuse_a, reuse_b)
      c = __builtin_amdgcn_wmma_f32_16x16x4_f32(false, aF[ch], false, bF[ch],
                                                (short)0, c, false, false);
    }
    return c;
  };

  float fm[8], fv[8], fs[8], fk[8];
  float acc = 0.0f;

  for (int ii = 0; ii < 2; ++ii) {
    const float* src = (ii == 0 ? pred : tgt) + pbase;
    __syncthreads();

    // ---- load raw tile + halo (zero padding to match explicit-pad conv) ----
    for (int idx = lane; idx < XS * XS; idx += 32) {
      const int r = idx / XS, c = idx % XS;
      const int gr = gr0 - H + r, gc = gc0 - H + c;
      float v = 0.0f;
      if (gr >= 0 && gr < IMG && gc >= 0 && gc < IMG) v = src[gr * IMG + gc];
      Xb[w][r * XSP + c] = v;
    }
    __syncthreads();

    // ---- mean: horizontal pass over full halo region (VALU) ----
    for (int idx = lane; idx < XS * MS; idx += 32) {
      const int r = idx / MS, c = idx % MS;
      float s = 0.0f;
#pragma unroll
      for (int j = 0; j < K; ++j) s += greg[j] * Xb[w][r * XSP + c + j];
      Tb[w][r * MSP + c] = s;
    }
    __syncthreads();

    // ---- mean: vertical pass -> Mb over (16+2P)^2 region (VALU) ----
    for (int idx = lane; idx < MS * MS; idx += 32) {
      const int r = idx / MS, c = idx % MS;
      float s = 0.0f;
#pragma unroll
      for (int j = 0; j < K; ++j) s += greg[j] * Tb[w][(r + j) * MSP + c];
      Mb[w][r * MSP + c] = s;
    }
    __syncthreads();

    // ---- E[x^2]: horizontal pass, center 16 columns ----
    for (int idx = lane; idx < MS * 16; idx += 32) {
      const int t = idx >> 4, n = idx & 15;
      float s = 0.0f;
#pragma unroll
      for (int j = 0; j < K; ++j) {
        const float x = Xb[w][(P + t) * XSP + (P + n + j)];
        s += greg[j] * x * x;
      }
      Tb[w][(P + t) * MSP + n] = s;
    }
    __syncthreads();
    const v8f cex2 = conv_v(&Tb[w][0]);   // vertical pass on matrix pipe

    // ---- m3: (x-mean)^3 then separable conv ----
    for (int idx = lane; idx < MS * MS; idx += 32) {
      const int r = idx / MS, c = idx % MS;
      const float xc = Xb[w][(P + r) * XSP + (P + c)] - Mb[w][r * MSP + c];
      Pb[w][r * MSP + c] = xc * xc * xc;
    }
    __syncthreads();
    for (int idx = lane; idx < MS * 16; idx += 32) {
      const int t = idx >> 4, n = idx & 15;
      float s = 0.0f;
#pragma unroll
      for (int j = 0; j < K; ++j) s += greg[j] * Pb[w][t * MSP + n + j];
      Tb[w][(P + t) * MSP + n] = s;
    }
    __syncthreads();
    const v8f cm3 = conv_v(&Tb[w][0]);

    // ---- m4: (x-mean)^4 then separable conv ----
    for (int idx = lane; idx < MS * MS; idx += 32) {
      const int r = idx / MS, c = idx % MS;
      const float xc = Xb[w][(P + r) * XSP + (P + c)] - Mb[w][r * MSP + c];
      const float x2 = xc * xc;
      Pb[w][r * MSP + c] = x2 * x2;
    }
    __syncthreads();
    for (int idx = lane; idx < MS * 16; idx += 32) {
      const int t = idx >> 4, n = idx & 15;
      float s = 0.0f;
#pragma unroll
      for (int j = 0; j < K; ++j) s += greg[j] * Pb[w][t * MSP + n + j];
      Tb[w][(P + t) * MSP + n] = s;
    }
    __syncthreads();
    const v8f cm4 = conv_v(&Tb[w][0]);

    // ---- features (WMMA C layout: row v+8*hi, col lo) ----
#pragma unroll
    for (int v = 0; v < 8; ++v) {
      const int m = v + 8 * hi;
      const float mean = Mb[w][(P + m) * MSP + (P + lo)];
      const float var  = fmaxf(cex2[v] - mean * mean, EPSF);
      const float sd   = sqrtf(var);
      const float skew = cm3[v] / (sd * sd * sd + EPSF);
      const float kurt = cm4[v] / (var * var + EPSF);
      if (ii == 0) {
        fm[v] = mean; fv[v] = var; fs[v] = skew; fk[v] = kurt;
      } else {
        acc += fabsf(mean - fm[v]) + fabsf(var - fv[v]) +
               0.5f * fabsf(skew - fs[v]) + 0.001f * fabsf(kurt - fk[v]);
      }
    }
  }

  // ---- wave32 reduction, then block partial (deterministic, no atomics) ----
#pragma unroll
  for (int off = 16; off > 0; off >>= 1) acc += __shfl_down(acc, off, 32);
  if (lane == 0) wsum[w] = acc;
  __syncthreads();
  if (tid == 0)
    partial[blockIdx.x] = wsum[0] + wsum[1] + wsum[2] + wsum[3];
}

__global__ __launch_bounds__(256) void sfl_reduce(
    const float* __restrict__ partial, int n, float* __restrict__ out) {
  __shared__ float sm[256];
  float s = 0.0f;
  for (int i = threadIdx.x; i < n; i += 256) s += partial[i];
  sm[threadIdx.x] = s;
  __syncthreads();
  for (int st = 128; st > 0; st >>= 1) {
    if (threadIdx.x < st) sm[threadIdx.x] += sm[threadIdx.x + st];
    __syncthreads();
  }
  if (threadIdx.x == 0)
    out[0] = sm[0] * (1.0f / (12.0f * (float)(NPLANES * IMG * IMG)));
}

extern "C" void kernel_launch(void* const* d_in, const int* in_sizes, int n_in,
                              void* d_out, int out_size, void* d_ws,
                              size_t ws_size, hipStream_t stream) {
  (void)in_sizes; (void)n_in; (void)out_size; (void)ws_size;
  const float* pred = (const float*)d_in[0];
  const float* tgt  = (const float*)d_in[1];
  float* out = (float*)d_out;
  float* ws  = (float*)d_ws;   // 3 * 12288 partials = 147 KB scratch

  dim3 blk(128);
  sfl_kernel<3><<<BLOCKS_PER_K, blk, 0, stream>>>(pred, tgt, ws);
  sfl_kernel<5><<<BLOCKS_PER_K, blk, 0, stream>>>(pred, tgt, ws + BLOCKS_PER_K);
  sfl_kernel<7><<<BLOCKS_PER_K, blk, 0, stream>>>(pred, tgt,
                                                  ws + 2 * BLOCKS_PER_K);
  sfl_reduce<<<1, 256, 0, stream>>>(ws, 3 * BLOCKS_PER_K, out);
}